// DetectionLoss_70592082477135
// MI455X (gfx1250) — compile-verified
//
#include <hip/hip_runtime.h>
#include <hip/hip_bf16.h>

typedef __attribute__((ext_vector_type(2))) float v2f;
typedef __attribute__((ext_vector_type(8))) float v8f;

#define DET_EPS 1e-6f
#define POS_THR 0.5f
#define ALPHA_C 0.5f
#define TG 4        // targets per k1 block
#define ITEMS 4     // anchors per thread in k2 (block=256 -> 1024 anchors/block)

// ---------------------------------------------------------------- k0: zero pack
__global__ void k0_zero(unsigned long long* __restrict__ pack, int n) {
  int i = blockIdx.x * blockDim.x + threadIdx.x;
  if (i < n) pack[i] = 0ull;
}

// ---------------------------------------------------------------- k1: best anchor per target
// grid(chunks, T/TG, B), block 256. Packed u64 atomicMax -> argmax over A with
// ties resolved to the smallest anchor index (matches jnp.argmax).
__global__ void k1_best_anchor(const float4* __restrict__ anchors,
                               const float4* __restrict__ tboxes,
                               unsigned long long* __restrict__ pack,
                               int A, int T, int anchorsPerChunk) {
  const int b   = blockIdx.z;
  const int t0  = blockIdx.y * TG;
  const int tid = threadIdx.x;

  float4 tb[TG]; float tarea[TG];
  #pragma unroll
  for (int j = 0; j < TG; ++j) {
    float4 x = tboxes[(size_t)b * T + t0 + j];
    tb[j] = x;
    tarea[j] = (x.z - x.x) * (x.w - x.y);
  }
  // track best as fraction (ni/nu) to avoid per-pair division
  float ni[TG], nu[TG]; int na[TG];
  #pragma unroll
  for (int j = 0; j < TG; ++j) { ni[j] = -1.0f; nu[j] = 1.0f; na[j] = 0; }

  const int aBase = blockIdx.x * anchorsPerChunk;
  for (int i = tid; i < anchorsPerChunk; i += 256) {
    const int a = aBase + i;
    float4 ab = anchors[a];
    float areaA = (ab.z - ab.x) * (ab.w - ab.y);
    #pragma unroll
    for (int j = 0; j < TG; ++j) {
      float x1 = fmaxf(ab.x, tb[j].x), y1 = fmaxf(ab.y, tb[j].y);
      float x2 = fminf(ab.z, tb[j].z), y2 = fminf(ab.w, tb[j].w);
      float inter = fmaxf(x2 - x1, 0.0f) * fmaxf(y2 - y1, 0.0f);
      float den = areaA + tarea[j] - inter + DET_EPS;   // union + eps > 0
      // iou_new > iou_best  <=>  inter*nu > ni*den  (scan is ascending in a,
      // strict > keeps the first/lowest index on ties)
      if (inter * nu[j] > ni[j] * den) { ni[j] = inter; nu[j] = den; na[j] = a; }
    }
  }

  __shared__ float sni[256], snu[256];
  __shared__ int   sna[256];
  for (int j = 0; j < TG; ++j) {
    sni[tid] = ni[j]; snu[tid] = nu[j]; sna[tid] = na[j];
    __syncthreads();
    for (int s = 128; s > 0; s >>= 1) {     // fixed-order tree, tie -> min index
      if (tid < s) {
        float i1 = sni[tid],     u1 = snu[tid];     int a1 = sna[tid];
        float i2 = sni[tid + s], u2 = snu[tid + s]; int a2 = sna[tid + s];
        float c1 = i1 * u2, c2 = i2 * u1;
        bool take2 = (c2 > c1) || ((c2 == c1) && (a2 < a1));
        if (take2) { sni[tid] = i2; snu[tid] = u2; sna[tid] = a2; }
      }
      __syncthreads();
    }
    if (tid == 0) {
      float iou = fmaxf(sni[0], 0.0f) / snu[0];   // non-negative -> bits monotone
      unsigned long long key =
          ((unsigned long long)__float_as_uint(iou) << 32) |
          (unsigned long long)(unsigned)(~sna[0]); // tie on iou -> smaller index wins
      atomicMax(&pack[(size_t)b * T + t0 + j], key);
    }
    __syncthreads();
  }
}

// ---------------------------------------------------------------- k2: per-anchor losses
// grid(PBLK, B), block 256. Writes one float4 partial (conf, ct, bbox, cls) per block.
__global__ void k2_main(const float4* __restrict__ bbox_pred,
                        const float*  __restrict__ conf,
                        const float*  __restrict__ cls,
                        const float4* __restrict__ anchors,
                        const float4* __restrict__ tboxes,
                        const int*    __restrict__ tlabels,
                        const unsigned long long* __restrict__ pack,
                        float4* __restrict__ partials,
                        int A, int T, int PBLK) {
  const int b   = blockIdx.y;
  const int tid = threadIdx.x;

  __shared__ float4 stb[64];
  __shared__ float  sarea[64];
  __shared__ int    slab[64];
  __shared__ int    sba[64];
  if (tid < T) {
    float4 x = tboxes[(size_t)b * T + tid];
    stb[tid] = x;
    sarea[tid] = (x.z - x.x) * (x.w - x.y);
    slab[tid] = tlabels[(size_t)b * T + tid];
    sba[tid]  = (int)~(unsigned)(pack[(size_t)b * T + tid] & 0xFFFFFFFFull);
  }
  __syncthreads();

  float accConf = 0.0f, accCt = 0.0f, accBox = 0.0f, accCls = 0.0f;
  const int aBase = blockIdx.x * (256 * ITEMS);

  for (int it = 0; it < ITEMS; ++it) {
    const int a = aBase + it * 256 + tid;
    // prefetch the next conf cacheline slice (speculative, dropped if OOB)
    __builtin_prefetch(&conf[(size_t)b * A + aBase + (it + 1) * 256 + tid], 0, 0);

    float4 ab = anchors[a];
    float areaA = (ab.z - ab.x) * (ab.w - ab.y);

    // argmax over targets, division-free
    float bi = -1.0f, bu = 1.0f; int bt = 0;
    #pragma unroll 8
    for (int t = 0; t < 64; ++t) {
      float4 tb = stb[t];
      float x1 = fmaxf(ab.x, tb.x), y1 = fmaxf(ab.y, tb.y);
      float x2 = fminf(ab.z, tb.z), y2 = fminf(ab.w, tb.w);
      float inter = fmaxf(x2 - x1, 0.0f) * fmaxf(y2 - y1, 0.0f);
      float den = areaA + sarea[t] - inter + DET_EPS;
      if (inter * bu > bi * den) { bi = inter; bu = den; bt = t; }
    }
    bool pos = (bi >= POS_THR * bu);           // iou >= 0.5
    #pragma unroll 8
    for (int t = 0; t < 64; ++t) pos = pos || (sba[t] == a);
    float ct = pos ? 1.0f : 0.0f;

    // ---- focal confidence loss (all anchors)
    float p = conf[(size_t)b * A + a];
    float alpha = pos ? (1.0f - ALPHA_C) : ALPHA_C;
    float w = (1.0f - p) * ct + p * (1.0f - ct);
    float focal = w * __fsqrt_rn(w);           // w^1.5
    float bce = -(ct * __logf(p + DET_EPS) + (1.0f - ct) * __logf(1.0f - p + DET_EPS));
    accConf += bce * focal * alpha;
    accCt   += ct;

    // ---- bbox + cls only touch memory for positives (ct gates them anyway)
    if (pos) {
      float4 bp = bbox_pred[(size_t)b * A + a];
      float4 mb = stb[bt];
      float x1 = fmaxf(bp.x, mb.x), y1 = fmaxf(bp.y, mb.y);
      float x2 = fminf(bp.z, mb.z), y2 = fminf(bp.w, mb.w);
      float inter = fmaxf(x2 - x1, 0.0f) * fmaxf(y2 - y1, 0.0f);
      float a1 = (bp.z - bp.x) * (bp.w - bp.y);
      float a2 = (mb.z - mb.x) * (mb.w - mb.y);
      float un = a1 + a2 - inter;
      float iou = inter / (un + DET_EPS);
      float ex1 = fminf(bp.x, mb.x), ey1 = fminf(bp.y, mb.y);
      float ex2 = fmaxf(bp.z, mb.z), ey2 = fmaxf(bp.w, mb.w);
      float enc = (ex2 - ex1) * (ey2 - ey1);
      float giou = iou - (enc - un) / (enc + DET_EPS);
      float l1 = (fabsf(bp.x - mb.x) + fabsf(bp.y - mb.y) +
                  fabsf(bp.z - mb.z) + fabsf(bp.w - mb.w)) * 0.25f;
      accBox += (1.0f - giou) + 0.5f * l1;

      // stable BCE-with-logits vs one-hot: sum(max(x,0)+log1p(exp(-|x|))) - x[label]
      const float4* row = (const float4*)(cls + ((size_t)b * A + a) * 20); // rows 80B -> 16B aligned
      int lab = slab[bt] - 1;
      float s = 0.0f, xl = 0.0f;
      #pragma unroll
      for (int g = 0; g < 5; ++g) {
        float4 v = row[g];
        float xs[4] = {v.x, v.y, v.z, v.w};
        #pragma unroll
        for (int u = 0; u < 4; ++u) {
          float x = xs[u];
          s += fmaxf(x, 0.0f) + log1pf(__expf(-fabsf(x)));
          if (g * 4 + u == lab) xl = x;
        }
      }
      accCls += (s - xl) * (1.0f / 20.0f);
    }
  }

  // fixed-order block reduction (deterministic)
  __shared__ float4 red[256];
  red[tid] = make_float4(accConf, accCt, accBox, accCls);
  __syncthreads();
  for (int s = 128; s > 0; s >>= 1) {
    if (tid < s) {
      float4 o = red[tid], r = red[tid + s];
      o.x += r.x; o.y += r.y; o.z += r.z; o.w += r.w;
      red[tid] = o;
    }
    __syncthreads();
  }
  if (tid == 0) partials[(size_t)b * PBLK + blockIdx.x] = red[0];
}

// ---------------------------------------------------------------- k3: WMMA final reduce
// One wave (32 lanes). A = all-ones 16x4 f32 => D[m,n] = C + sum_k B[k,n]:
// exact f32 column sums; column n = image index. Chain K over all PBLK partials.
__global__ void k3_final(const float4* __restrict__ partials,
                         int B, int PBLK, int A, float* __restrict__ out) {
  const int lane = threadIdx.x;
  __shared__ float sums[4][16];

  v2f ones; ones.x = 1.0f; ones.y = 1.0f;
  const int n    = lane & 15;
  const int koff = (lane >> 4) * 2;
  const float* pf = (const float*)partials;

  for (int q = 0; q < 4; ++q) {
    v8f acc = {};
    for (int p0 = 0; p0 < PBLK; p0 += 4) {
      const int k0 = p0 + koff;
      v2f bv;
      bv.x = pf[((size_t)n * PBLK + k0) * 4 + q];
      bv.y = pf[((size_t)n * PBLK + k0 + 1) * 4 + q];
      acc = __builtin_amdgcn_wmma_f32_16x16x4_f32(
          /*neg_a=*/false, ones, /*neg_b=*/false, bv,
          /*c_mod=*/(short)0, acc, /*reuse_a=*/false, /*reuse_b=*/false);
    }
    if (lane < 16) sums[q][lane] = acc[0];  // every row of D equals the column sum
  }
  __syncthreads();

  if (lane == 0) {
    float tc = 0.0f, tb = 0.0f, tcl = 0.0f;
    for (int b = 0; b < B; ++b) {
      float confb = sums[0][b] / (float)A;
      float npos  = fmaxf(sums[1][b], 1.0f);
      float boxb  = sums[2][b] / npos;
      float clsb  = sums[3][b] / npos;
      tc += confb; tb += boxb; tcl += clsb;
    }
    float invB = 1.0f / (float)B;
    tc *= invB; tb *= invB; tcl *= invB;
    out[0] = tc + tb + tcl;
    out[1] = tc;
    out[2] = tb;
    out[3] = tcl;
  }
}

// ---------------------------------------------------------------- launch
extern "C" void kernel_launch(void* const* d_in, const int* in_sizes, int n_in,
                              void* d_out, int out_size, void* d_ws, size_t ws_size,
                              hipStream_t stream) {
  const float4* bbox_pred = (const float4*)d_in[0];
  const float*  conf_pred = (const float*)d_in[1];
  const float*  cls_pred  = (const float*)d_in[2];
  const float4* anchors   = (const float4*)d_in[3];
  const float4* tboxes    = (const float4*)d_in[4];
  const int*    tlabels   = (const int*)d_in[5];
  float* out = (float*)d_out;

  const int A = in_sizes[3] / 4;        // 65536
  const int B = in_sizes[1] / A;        // 16
  const int T = in_sizes[5] / B;        // 64

  unsigned long long* pack = (unsigned long long*)d_ws;                 // B*T u64
  float4* partials = (float4*)((char*)d_ws + ((size_t)B * T * 8 + 255 & ~(size_t)255));

  const int anchorsPerBlock = 256 * ITEMS;      // 1024
  const int PBLK = A / anchorsPerBlock;         // 64 (multiple of 4)
  const int chunks = 8;
  const int anchorsPerChunk = A / chunks;       // 8192

  // 1) zero the packed best-anchor table (ws is poisoned between runs)
  {
    int n = B * T;
    k0_zero<<<(n + 255) / 256, 256, 0, stream>>>(pack, n);
  }
  // 2) global argmax-over-anchors per target
  {
    dim3 grid(chunks, T / TG, B);
    k1_best_anchor<<<grid, 256, 0, stream>>>(anchors, tboxes, pack, A, T, anchorsPerChunk);
  }
  // 3) per-anchor losses -> per-block partials
  {
    dim3 grid(PBLK, B);
    k2_main<<<grid, 256, 0, stream>>>(bbox_pred, conf_pred, cls_pred, anchors,
                                      tboxes, tlabels, pack, partials, A, T, PBLK);
  }
  // 4) WMMA-based exact final reduction + finalize
  k3_final<<<1, 32, 0, stream>>>(partials, B, PBLK, A, out);
}